// InferenceLayer_532575945009
// MI455X (gfx1250) — compile-verified
//
#include <hip/hip_runtime.h>
#include <math.h>

typedef __attribute__((ext_vector_type(2))) float v2f;
typedef __attribute__((ext_vector_type(4))) float v4f;
typedef __attribute__((ext_vector_type(8))) float v8f;

#define NB   64
#define NT   512
#define ND   1024
#define NK   32
#define DCHUNK 512   // stage W in 64KB LDS chunks (K-paired, transposed)

// ---------------------------------------------------------------------------
// Kernel 1: logits[b,t,k] = sum_d vectors[bt,d] * W[k,d] + bias[k]
// M = B*T = 32768, N = K = 32, inner = D = 1024.  Memory-bound (~15 FLOP/B),
// so native fp32 WMMA (v_wmma_f32_16x16x4_f32) is the right matrix path.
// One wave -> 16(M) x 32(N) tile via two 16x16 accumulators.
//
// LDS layout (K-paired): element W[k][d] -> Wt2[((d>>1)*NK + k)*2 + (d&1)].
// A lane's B-fragment rows (k, k+1) are then 8B-contiguous, so each fragment
// is a single ds_load_b64 straight into an even-aligned VGPR pair (no packing
// v_movs before the WMMA).
// ---------------------------------------------------------------------------
__global__ __launch_bounds__(256)
void gemm_logits_kernel(const float* __restrict__ V,
                        const float* __restrict__ W,
                        const float* __restrict__ bias,
                        float* __restrict__ logits)
{
    __shared__ float Wt2[DCHUNK * NK];   // 64 KB

    const int tid  = threadIdx.x;
    const int wave = tid >> 5;
    const int lane = tid & 31;
    const int half = lane >> 4;     // 0: K pair (d0,d0+1), 1: (d0+2,d0+3)
    const int l15  = lane & 15;
    const int mbase = (blockIdx.x * 8 + wave) * 16;

    // A-fragment source row for this lane (16x4 fp32 layout, ISA 7.12.2)
    const float* arow = V + (size_t)(mbase + l15) * ND + half * 2;

    v8f acc0, acc1;
    const float b0 = bias[l15];
    const float b1 = bias[16 + l15];
#pragma unroll
    for (int v = 0; v < 8; ++v) { acc0[v] = b0; acc1[v] = b1; }

    for (int dchunk = 0; dchunk < ND; dchunk += DCHUNK) {
        // hint the L2 about the next chunk of the A stream (the 134MB
        // roofline-critical traffic) -> global_prefetch_b8
        if (dchunk + DCHUNK < ND)
            __builtin_prefetch(arow + dchunk + DCHUNK, 0, 1);

        __syncthreads();
        // stage W[k][dchunk..+511] into K-paired LDS: coalesced b128 reads
        for (int e = tid * 4; e < DCHUNK * NK; e += 256 * 4) {
            int k  = e >> 9;              // / DCHUNK
            int dd = e & (DCHUNK - 1);
            v4f w4 = *(const v4f*)(W + (size_t)k * ND + dchunk + dd);
#pragma unroll
            for (int q = 0; q < 4; ++q) {
                int d = dd + q;
                Wt2[((d >> 1) * NK + k) * 2 + (d & 1)] = w4[q];
            }
        }
        __syncthreads();

#pragma unroll 4
        for (int d0 = 0; d0 < DCHUNK; d0 += 4) {
            // A: lanes0-15 rows m, K=(d0,d0+1); lanes16-31 K=(d0+2,d0+3)
            v2f a = *(const v2f*)(arow + dchunk + d0);
            // B (4x16): lane reads rows (k,k+1) for col n as one b64
            const int prow = ((d0 >> 1) + half) * (NK * 2);
            v2f bw0 = *(const v2f*)(&Wt2[prow + l15 * 2]);
            v2f bw1 = *(const v2f*)(&Wt2[prow + (16 + l15) * 2]);
            acc0 = __builtin_amdgcn_wmma_f32_16x16x4_f32(
                false, a, false, bw0, (short)0, acc0, false, false);
            acc1 = __builtin_amdgcn_wmma_f32_16x16x4_f32(
                false, a, false, bw1, (short)0, acc1, false, false);
        }
    }

    // C/D layout: VGPR v -> row (v + 8*half), column = l15 (+16 for acc1)
    float* orow = logits + (size_t)(mbase + half * 8) * NK + l15;
#pragma unroll
    for (int v = 0; v < 8; ++v) {
        orow[(size_t)v * NK]      = acc0[v];
        orow[(size_t)v * NK + 16] = acc1[v];
    }
}

// ---------------------------------------------------------------------------
// Kernel 2: CRF forward algorithm + gold score, one wave32 per batch element.
// Lane j owns alpha[j]; transition column trans[.][j] cached in 32 VGPRs;
// alpha_i broadcast via v_readlane (wave-uniform -> SALU source).
// ---------------------------------------------------------------------------
__global__ __launch_bounds__(32)
void crf_kernel(const float* __restrict__ logits,
                const int*   __restrict__ mask,
                const int*   __restrict__ targets,
                const float* __restrict__ trans,
                const float* __restrict__ start_t,
                const float* __restrict__ end_t,
                float* __restrict__ ws)
{
    const int b = blockIdx.x;
    const int j = threadIdx.x;
    const float* lg = logits + (size_t)b * NT * NK;
    const int*   mk = mask    + b * NT;
    const int*   tg = targets + b * NT;

    float tc[NK];
#pragma unroll
    for (int i = 0; i < NK; ++i) tc[i] = trans[i * NK + j];

    float alpha = start_t[j] + lg[j];

    for (int t = 1; t < NT; ++t) {
        const float lj = lg[t * NK + j];
        const int   mt = mk[t];
        // logsumexp over i of (alpha_i + trans[i][j])
        float m = -3.402823466e38f;
#pragma unroll
        for (int i = 0; i < NK; ++i) {
            float ai = __int_as_float(
                __builtin_amdgcn_readlane(__float_as_int(alpha), i));
            m = fmaxf(m, ai + tc[i]);
        }
        float s = 0.f;
#pragma unroll
        for (int i = 0; i < NK; ++i) {
            float ai = __int_as_float(
                __builtin_amdgcn_readlane(__float_as_int(alpha), i));
            s += __expf(ai + tc[i] - m);
        }
        const float na = m + __logf(s) + lj;
        alpha = (mt > 0) ? na : alpha;
    }

    // partition = logsumexp_j(alpha_j + end_trans_j): butterfly reduce
    float x = alpha + end_t[j];
    float m = x;
#pragma unroll
    for (int off = 16; off; off >>= 1) m = fmaxf(m, __shfl_xor(m, off, 32));
    float e = __expf(x - m);
#pragma unroll
    for (int off = 16; off; off >>= 1) e += __shfl_xor(e, off, 32);
    const float partition = m + __logf(e);

    // gold score: lanes stride over time steps
    float emit_sum = 0.f, trans_sum = 0.f;
    int msum = 0;
    for (int t = j; t < NT; t += 32) {
        const float mf = (float)mk[t];
        emit_sum += lg[t * NK + tg[t]] * mf;
        if (t >= 1) trans_sum += trans[tg[t - 1] * NK + tg[t]] * mf;
        msum += mk[t];
    }
#pragma unroll
    for (int off = 16; off; off >>= 1) {
        emit_sum  += __shfl_xor(emit_sum,  off, 32);
        trans_sum += __shfl_xor(trans_sum, off, 32);
        msum      += __shfl_xor(msum,      off, 32);
    }

    if (j == 0) {
        const int last_tag = tg[msum - 1];
        const float score = start_t[tg[0]] + emit_sum + trans_sum + end_t[last_tag];
        ws[b]      = score - partition;
        ws[NB + b] = (float)msum;
    }
}

// ---------------------------------------------------------------------------
// Kernel 3: fold 64 per-batch results into the scalar loss.
// ---------------------------------------------------------------------------
__global__ __launch_bounds__(64)
void finalize_kernel(const float* __restrict__ ws, float* __restrict__ out)
{
    __shared__ float ssc[NB], smc[NB];
    const int t = threadIdx.x;
    ssc[t] = ws[t];
    smc[t] = ws[NB + t];
    __syncthreads();
    if (t == 0) {
        float a = 0.f, mm = 0.f;
        for (int i = 0; i < NB; ++i) { a += ssc[i]; mm += smc[i]; }
        out[0] = -a / mm;
    }
}

extern "C" void kernel_launch(void* const* d_in, const int* in_sizes, int n_in,
                              void* d_out, int out_size, void* d_ws, size_t ws_size,
                              hipStream_t stream) {
    const float* vectors  = (const float*)d_in[0];
    const int*   mask     = (const int*)  d_in[1];
    const int*   targets  = (const int*)  d_in[2];
    const float* W        = (const float*)d_in[3];
    const float* bias     = (const float*)d_in[4];
    const float* trans    = (const float*)d_in[5];
    const float* start_t  = (const float*)d_in[6];
    const float* end_t    = (const float*)d_in[7];

    float* out    = (float*)d_out;
    float* logits = out + 1;          // tuple output: [loss, logits(B*T*K)]
    float* ws     = (float*)d_ws;

    gemm_logits_kernel<<<256, 256, 0, stream>>>(vectors, W, bias, logits);
    crf_kernel<<<NB, 32, 0, stream>>>(logits, mask, targets, trans,
                                      start_t, end_t, ws);
    finalize_kernel<<<1, 64, 0, stream>>>(ws, out);
}